// CenterLoss_9448928051410
// MI455X (gfx1250) — compile-verified
//
#include <hip/hip_runtime.h>

// Center loss, algebraically reduced to the diagonal of the distance matrix:
//   loss = (1/B) * sum_b clip(||x_b - c_{labels[b]}||^2, 1e-12, 1e12)
//          + (C-1)*1e-12          // clamped zeros of the masked matrix
//
// Bandwidth-bound: ~16.8 MB of traffic vs 42 GFLOP for the naive GEMM.
// Dot products go through V_WMMA_F32_16X16X4_F32 (full fp32 precision):
// one wave32 per 16 batch rows, K=256 accumulated over 64 WMMAs.

typedef __attribute__((ext_vector_type(2))) float v2f;
typedef __attribute__((ext_vector_type(8))) float v8f;

#define BATCH       8192
#define NUM_CLASSES 10000
#define FEAT_DIM    256
#define ROWS_PER_WAVE 16
#define NUM_TILES   (BATCH / ROWS_PER_WAVE)   // 512 waves

__global__ __launch_bounds__(32)
void center_loss_tile_wmma(const float* __restrict__ x,
                           const int*   __restrict__ labels,
                           const float* __restrict__ centers,
                           float*       __restrict__ partials) {
  const int lane  = threadIdx.x;        // 0..31, wave32
  const int n16   = lane & 15;          // M (A) / N (B) index within the tile
  const int khalf = (lane >> 4) * 2;    // K sub-offset: lanes 0-15 -> K{0,1}, 16-31 -> K{2,3}

  const int row = blockIdx.x * ROWS_PER_WAVE + n16;
  const int lab = labels[row];          // gather which center this row compares against

  const float* __restrict__ xr = x       + (size_t)row * FEAT_DIM + khalf;
  const float* __restrict__ cr = centers + (size_t)lab * FEAT_DIM + khalf;

  v8f acc0 = {};      // 16x16 f32 accumulators (2 chains to break WMMA RAW dependency)
  v8f acc1 = {};
  float xn = 0.f;     // partial ||x_row||^2 over this lane's K-half
  float cn = 0.f;     // partial ||c_lab||^2 over this lane's K-half

#pragma unroll
  for (int k = 0; k < FEAT_DIM; k += 8) {
    v2f a0 = *(const v2f*)(xr + k);       // A[m][k..k+1] for this lane's K-half
    v2f b0 = *(const v2f*)(cr + k);       // B[k..k+1][n] = centers[label_n][k..]
    v2f a1 = *(const v2f*)(xr + k + 4);
    v2f b1 = *(const v2f*)(cr + k + 4);
    acc0 = __builtin_amdgcn_wmma_f32_16x16x4_f32(
        /*neg_a=*/false, a0, /*neg_b=*/false, b0,
        /*c_mod=*/(short)0, acc0, /*reuse_a=*/false, /*reuse_b=*/false);
    acc1 = __builtin_amdgcn_wmma_f32_16x16x4_f32(
        false, a1, false, b1, (short)0, acc1, false, false);
    xn += a0[0]*a0[0] + a0[1]*a0[1] + a1[0]*a1[0] + a1[1]*a1[1];
    cn += b0[0]*b0[0] + b0[1]*b0[1] + b1[0]*b1[0] + b1[1]*b1[1];
  }

  v8f dsum = acc0 + acc1;

  // Combine the two K-halves of the norms (lane L <-> lane L^16 hold the same row).
  float xs = xn + __shfl_xor(xn, 16, 32);
  float cs = cn + __shfl_xor(cn, 16, 32);

  // Extract the diagonal D[m][m] of the 16x16 product.
  // C/D layout: lanes 0-15 hold N=lane with VGPR r = row M=r;
  //             lanes 16-31 hold N=lane-16 with VGPR r = row M=r+8.
  // => diag m<8 at lane m, VGPR m; diag m>=8 at lane m+16, VGPR m-8.
  const int s = lane & 7;
  float dot = 0.f;
  switch (s) {
    case 0: dot = dsum[0]; break;
    case 1: dot = dsum[1]; break;
    case 2: dot = dsum[2]; break;
    case 3: dot = dsum[3]; break;
    case 4: dot = dsum[4]; break;
    case 5: dot = dsum[5]; break;
    case 6: dot = dsum[6]; break;
    case 7: dot = dsum[7]; break;
  }
  const bool valid = (lane < 8) || (lane >= 24);

  float dist = xs + cs - 2.0f * dot;            // ||x||^2 + ||c||^2 - 2<x,c>
  dist = fminf(fmaxf(dist, 1e-12f), 1e12f);     // torch-style clamp
  float v = valid ? dist : 0.f;

  // Wave32 butterfly sum of the 16 clamped distances.
#pragma unroll
  for (int off = 16; off > 0; off >>= 1) v += __shfl_xor(v, off, 32);

  if (lane == 0) partials[blockIdx.x] = v;
}

__global__ __launch_bounds__(NUM_TILES)
void center_loss_reduce(const float* __restrict__ partials,
                        float*       __restrict__ out) {
  __shared__ float smem[NUM_TILES];
  const int t = threadIdx.x;
  smem[t] = partials[t];
  __syncthreads();
#pragma unroll
  for (int off = NUM_TILES / 2; off > 0; off >>= 1) {
    if (t < off) smem[t] += smem[t + off];
    __syncthreads();
  }
  if (t == 0) {
    // + (C-1)*1e-12: the B*(C-1) masked zeros each clamp up to 1e-12, /B.
    out[0] = smem[0] / (float)BATCH + (float)(NUM_CLASSES - 1) * 1e-12f;
  }
}

extern "C" void kernel_launch(void* const* d_in, const int* in_sizes, int n_in,
                              void* d_out, int out_size, void* d_ws, size_t ws_size,
                              hipStream_t stream) {
  const float* x       = (const float*)d_in[0];   // [8192, 256] f32
  const int*   labels  = (const int*)  d_in[1];   // [8192] int
  const float* centers = (const float*)d_in[2];   // [10000, 256] f32
  float*       out     = (float*)d_out;           // scalar
  float*       partials = (float*)d_ws;           // 512 floats of scratch

  center_loss_tile_wmma<<<NUM_TILES, 32, 0, stream>>>(x, labels, centers, partials);
  center_loss_reduce<<<1, NUM_TILES, 0, stream>>>(partials, out);
}